// CrossAttention_45286135169187
// MI455X (gfx1250) — compile-verified
//
#include <hip/hip_runtime.h>

// Problem constants (fixed by the reference): B=16, DIM=512, H=W=64, HW=4096.
#define NB   16
#define DIM  512
#define HW   4096

typedef float v2f __attribute__((ext_vector_type(2)));
typedef float v8f __attribute__((ext_vector_type(8)));

// ---------------------------------------------------------------------------
// Stage 1: tq[b,x] = target[b,x]*w_q + b_q ; T[b] = sum_x tq[b,x]
// One block per batch, 256 threads, block-level tree reduce.
// ---------------------------------------------------------------------------
__global__ void k_tq(const float* __restrict__ target, const float* __restrict__ wq,
                     const float* __restrict__ bq, float* __restrict__ tq,
                     float* __restrict__ T) {
  __shared__ float red[256];
  const int b   = blockIdx.x;
  const int tid = threadIdx.x;
  const float w = wq[0];
  const float c = bq[0];
  float s = 0.0f;
#pragma unroll
  for (int i = 0; i < HW / 256; ++i) {
    const int x = tid + i * 256;
    const float v = target[b * HW + x] * w + c;
    tq[b * HW + x] = v;
    s += v;
  }
  red[tid] = s;
  __syncthreads();
  for (int off = 128; off > 0; off >>= 1) {
    if (tid < off) red[tid] += red[tid + off];
    __syncthreads();
  }
  if (tid == 0) T[b] = red[0];
}

// ---------------------------------------------------------------------------
// Stage 2: r[b,c] = sum_x storage[b,c,x] * tq[b,x]
// One block per (b,c) row; fully coalesced stream of storage (134 MB).
// ---------------------------------------------------------------------------
__global__ void k_r(const float* __restrict__ storage, const float* __restrict__ tq,
                    float* __restrict__ r) {
  __shared__ float red[256];
  const int bc  = blockIdx.x;       // b*DIM + c
  const int b   = bc >> 9;
  const int tid = threadIdx.x;
  const float* __restrict__ srow = storage + (size_t)bc * HW;
  const float* __restrict__ trow = tq + (size_t)b * HW;
  float s = 0.0f;
#pragma unroll
  for (int i = 0; i < HW / 256; ++i) {
    const int x = tid + i * 256;
    s += srow[x] * trow[x];
  }
  red[tid] = s;
  __syncthreads();
  for (int off = 128; off > 0; off >>= 1) {
    if (tid < off) red[tid] += red[tid + off];
    __syncthreads();
  }
  if (tid == 0) r[bc] = red[0];
}

// ---------------------------------------------------------------------------
// Stage 3 (WMMA): scores[b,k] = sum_c w_ca[DIM+k,c]*r[b,c] + b_ca[DIM+k]*T[b]
// GEMM M=16 (batch), N=512 (k), K=512 (c) via V_WMMA_F32_16X16X4_F32.
// One wave (32 threads) per 16-wide N tile; 128 chained WMMAs over K.
// A layout (16x4 f32): lanes 0-15 -> M=lane, v0/v1 = K=0/1; lanes 16-31 -> K=2/3.
// B layout (4x16 f32): lane = (K>=2?16:0)+N, vgpr = K&1.
// C/D layout: lane<16 -> M=v, N=lane ; lane>=16 -> M=v+8, N=lane-16.
// ---------------------------------------------------------------------------
__global__ void k_scores(const float* __restrict__ r, const float* __restrict__ w_ca,
                         const float* __restrict__ b_ca, const float* __restrict__ T,
                         float* __restrict__ scores) {
  const int lane  = threadIdx.x;        // 0..31, one full wave, EXEC all ones
  const int n0    = blockIdx.x * 16;    // output-k tile base
  const int nn    = lane & 15;          // M row for A, N col for B
  const int khalf = lane >> 4;          // K-half select
  v8f acc = {};
  const float* __restrict__ arow = r + (size_t)nn * DIM + khalf * 2;
  const float* __restrict__ brow = w_ca + (size_t)(DIM + n0 + nn) * DIM + khalf * 2;
#pragma unroll 4
  for (int c = 0; c < DIM; c += 4) {
    const v2f a = *(const v2f*)(arow + c);
    const v2f bm = *(const v2f*)(brow + c);
    acc = __builtin_amdgcn_wmma_f32_16x16x4_f32(false, a, false, bm,
                                                (short)0, acc, false, false);
  }
  const int k = n0 + nn;
  const float bias = b_ca[DIM + k];
#pragma unroll
  for (int v = 0; v < 8; ++v) {
    const int bidx = (lane < 16) ? v : v + 8;
    scores[bidx * DIM + k] = acc[v] + bias * T[bidx];
  }
}

// ---------------------------------------------------------------------------
// Stage 4: softmax over k (512) per batch; beta[b] = sum_k attn[b,k]*b_ca[k]
// ---------------------------------------------------------------------------
__global__ void k_softmax(const float* __restrict__ scores, const float* __restrict__ b_ca,
                          float* __restrict__ attn, float* __restrict__ beta) {
  __shared__ float red[256];
  const int b   = blockIdx.x;
  const int tid = threadIdx.x;
  const float v0 = scores[b * DIM + tid];
  const float v1 = scores[b * DIM + tid + 256];
  // max reduce
  red[tid] = fmaxf(v0, v1);
  __syncthreads();
  for (int off = 128; off > 0; off >>= 1) {
    if (tid < off) red[tid] = fmaxf(red[tid], red[tid + off]);
    __syncthreads();
  }
  const float m = red[0];
  __syncthreads();
  const float e0 = __expf(v0 - m);
  const float e1 = __expf(v1 - m);
  // sum reduce
  red[tid] = e0 + e1;
  __syncthreads();
  for (int off = 128; off > 0; off >>= 1) {
    if (tid < off) red[tid] += red[tid + off];
    __syncthreads();
  }
  const float inv = 1.0f / red[0];
  __syncthreads();
  const float a0 = e0 * inv;
  const float a1 = e1 * inv;
  attn[b * DIM + tid]       = a0;
  attn[b * DIM + tid + 256] = a1;
  // beta reduce
  red[tid] = a0 * b_ca[tid] + a1 * b_ca[tid + 256];
  __syncthreads();
  for (int off = 128; off > 0; off >>= 1) {
    if (tid < off) red[tid] += red[tid + off];
    __syncthreads();
  }
  if (tid == 0) beta[b] = red[0];
}

// ---------------------------------------------------------------------------
// Stage 5 (WMMA): u[b,c] = sum_k attn[b,k] * w_ca[k,c]
// GEMM M=16 (batch), N=512 (c), K=512 (k); same wave/tile scheme as stage 3.
// B elements come from different w_ca rows per K -> two scalar loads.
// ---------------------------------------------------------------------------
__global__ void k_u(const float* __restrict__ attn, const float* __restrict__ w_ca,
                    float* __restrict__ u) {
  const int lane  = threadIdx.x;
  const int n0    = blockIdx.x * 16;    // c tile base
  const int nn    = lane & 15;
  const int khalf = lane >> 4;
  v8f acc = {};
  const float* __restrict__ arow = attn + (size_t)nn * DIM + khalf * 2;
  const float* __restrict__ bcol = w_ca + (size_t)(khalf * 2) * DIM + n0 + nn;
#pragma unroll 4
  for (int k = 0; k < DIM; k += 4) {
    const v2f a = *(const v2f*)(arow + k);
    v2f bm;
    bm.x = bcol[(size_t)k * DIM];
    bm.y = bcol[(size_t)(k + 1) * DIM];
    acc = __builtin_amdgcn_wmma_f32_16x16x4_f32(false, a, false, bm,
                                                (short)0, acc, false, false);
  }
#pragma unroll
  for (int v = 0; v < 8; ++v) {
    const int bidx = (lane < 16) ? v : v + 8;
    u[bidx * DIM + n0 + nn] = acc[v];
  }
}

// ---------------------------------------------------------------------------
// Stage 6: final[b,d,x] = sum_c u[b,c]*storage[b,c,x] + beta[b], broadcast d.
// Block = (batch, 256-wide x tile): streams storage coalesced, 512x broadcast
// store (134 MB write). This stage is the HBM roofline of the whole problem.
// ---------------------------------------------------------------------------
__global__ void k_final(const float* __restrict__ storage, const float* __restrict__ u,
                        const float* __restrict__ beta, float* __restrict__ out) {
  const int b = blockIdx.y;
  const int x = blockIdx.x * 256 + threadIdx.x;
  const float* __restrict__ sb = storage + (size_t)b * DIM * HW + x;
  const float* __restrict__ ub = u + (size_t)b * DIM;
  float acc = 0.0f;
#pragma unroll 8
  for (int c = 0; c < DIM; ++c) {
    acc += ub[c] * sb[(size_t)c * HW];
  }
  acc += beta[b];
  float* __restrict__ ob = out + (size_t)b * DIM * HW + x;
#pragma unroll 8
  for (int d = 0; d < DIM; ++d) {
    ob[(size_t)d * HW] = acc;
  }
}

// ---------------------------------------------------------------------------
extern "C" void kernel_launch(void* const* d_in, const int* in_sizes, int n_in,
                              void* d_out, int out_size, void* d_ws, size_t ws_size,
                              hipStream_t stream) {
  const float* storage = (const float*)d_in[0];   // (16,512,64,64)
  const float* target  = (const float*)d_in[1];   // (16,1,64,64)
  const float* w_ca    = (const float*)d_in[2];   // (1024,512)
  const float* b_ca    = (const float*)d_in[3];   // (1024,)
  const float* w_q     = (const float*)d_in[4];   // (1,1)
  const float* b_q     = (const float*)d_in[5];   // (1,)
  float* out = (float*)d_out;

  // Workspace layout (floats). All buffers fully written before read each call.
  float* ws     = (float*)d_ws;
  float* tq     = ws;                        // 16*4096 = 65536
  float* T      = tq + NB * HW;              // 16 (pad to 64)
  float* r      = T + 64;                    // 16*512 = 8192
  float* scores = r + NB * DIM;              // 8192
  float* attn   = scores + NB * DIM;         // 8192
  float* u      = attn + NB * DIM;           // 8192
  float* beta   = u + NB * DIM;              // 16

  k_tq     <<<NB, 256, 0, stream>>>(target, w_q, b_q, tq, T);
  k_r      <<<NB * DIM, 256, 0, stream>>>(storage, tq, r);
  k_scores <<<DIM / 16, 32, 0, stream>>>(r, w_ca, b_ca, T, scores);
  k_softmax<<<NB, 256, 0, stream>>>(scores, b_ca, attn, beta);
  k_u      <<<DIM / 16, 32, 0, stream>>>(attn, w_ca, u);
  k_final  <<<dim3(HW / 256, NB), 256, 0, stream>>>(storage, u, beta, out);
}